// AttentiveRouter_37623913513507
// MI455X (gfx1250) — compile-verified
//
#include <hip/hip_runtime.h>
#include <hip/hip_bf16.h>
#include <math.h>

// Problem constants (from reference): B=64, S=1024, D=1024, E=8, K=2
#define BATCH 64
#define SEQ   1024
#define DIM   1024
#define NEXP  8
#define NEXP_PAD 16
#define TOPK  2

// router_main tiling: 8 waves x 2 tiles x 16 rows = 256 rows/block
#define BK            64
#define NCHUNK        (DIM / BK)                         // 16
#define WAVES_PER_BLK 8
#define THREADS       (WAVES_PER_BLK * 32)               // 256
#define ROWS_PER_BLK  (WAVES_PER_BLK * 32)               // 256 (2 tiles/wave)
#define LDS_STRIDE    (BK + 4)                           // bank-conflict-free
#define NBLOCKS_MAIN  ((BATCH * SEQ) / ROWS_PER_BLK)     // 256
#define BLOCKS_PER_BATCH (SEQ / ROWS_PER_BLK)            // 4

#define MASK_ELEMS (BATCH * SEQ * NEXP)                  // 524288

typedef __attribute__((ext_vector_type(2))) float v2f;
typedef __attribute__((ext_vector_type(8))) float v8f;
// pointee type for the async-LDS builtin (matches toolchain signature exactly)
typedef int vsi4 __attribute__((vector_size(16)));

// CDNA5 async global->LDS path (ASYNCcnt-tracked, bypasses VGPRs)
#if defined(__gfx1250__) && __has_builtin(__builtin_amdgcn_global_load_async_to_lds_b128)
#define HAS_ASYNC_LDS 1
#else
#define HAS_ASYNC_LDS 0
#endif

__device__ __forceinline__ void async_wait0() {
#if HAS_ASYNC_LDS
#if __has_builtin(__builtin_amdgcn_s_wait_asynccnt)
    __builtin_amdgcn_s_wait_asynccnt(0);
#else
    asm volatile("s_wait_asynccnt 0x0" ::: "memory");
#endif
#endif
}

// ---------------------------------------------------------------------------
// Kernel 1: fold router weights.  We[e,d] = sum_o key_emb[e,o] * Wq[o,d]
//           be[e]   = sum_o key_emb[e,o] * bq[o]
// We stored padded to 16 rows (rows 8..15 zero).
// ---------------------------------------------------------------------------
__global__ __launch_bounds__(256) void router_prep(
    const float* __restrict__ Wq, const float* __restrict__ bq,
    const float* __restrict__ key_emb,
    float* __restrict__ We, float* __restrict__ be)
{
    const int d = blockIdx.x * 256 + threadIdx.x;   // 4 blocks cover D=1024
    float acc[NEXP];
#pragma unroll
    for (int e = 0; e < NEXP; ++e) acc[e] = 0.0f;

    for (int o = 0; o < DIM; ++o) {
        const float wq = Wq[(size_t)o * DIM + d];
#pragma unroll
        for (int e = 0; e < NEXP; ++e)
            acc[e] = fmaf(key_emb[e * DIM + o], wq, acc[e]);
    }
#pragma unroll
    for (int e = 0; e < NEXP; ++e) We[(size_t)e * DIM + d] = acc[e];
#pragma unroll
    for (int e = NEXP; e < NEXP_PAD; ++e) We[(size_t)e * DIM + d] = 0.0f;

    if (blockIdx.x == 0 && threadIdx.x < NEXP_PAD) {
        const int e = threadIdx.x;
        float a = 0.0f;
        if (e < NEXP) {
            for (int o = 0; o < DIM; ++o)
                a = fmaf(key_emb[e * DIM + o], bq[o], a);
        }
        be[e] = a;
    }
}

// ---------------------------------------------------------------------------
// Kernel 2: fused scores GEMM (WMMA f32 16x16x4) + softmax + deterministic
// per-block partial reduction.  Double-buffered async global->LDS staging.
// ---------------------------------------------------------------------------
__global__ __launch_bounds__(THREADS) void router_main(
    const float* __restrict__ x, const float* __restrict__ We,
    const float* __restrict__ be,
    float* __restrict__ block_scores,   // [NBLOCKS_MAIN][NEXP]
    float* __restrict__ block_aux)      // [NBLOCKS_MAIN]
{
    __shared__ float xs[2][ROWS_PER_BLK * LDS_STRIDE];
    __shared__ float wpart[WAVES_PER_BLK][NEXP_PAD];
    __shared__ float apart[WAVES_PER_BLK];

    const int tid  = threadIdx.x;
    const int wave = tid >> 5;
    const int lane = tid & 31;
    const int row0 = blockIdx.x * ROWS_PER_BLK;     // flat (b*S + s) base

    const int ecol  = lane & 15;                    // expert column (N)
    const int khalf = (lane >> 4) << 1;             // K offset: 0 or 2

    // stage one BK-wide chunk of 256 x-rows into LDS buffer `dstbuf`
    auto issue_chunk = [&](int chunk, float* dstbuf) {
        const int k0 = chunk * BK;
#pragma unroll
        for (int i = tid; i < ROWS_PER_BLK * (BK / 4); i += THREADS) {
            const int r  = i >> 4;
            const int c4 = (i & 15) << 2;
            const float* gp = x + (size_t)(row0 + r) * DIM + k0 + c4;
            float* lp = &dstbuf[r * LDS_STRIDE + c4];
#if HAS_ASYNC_LDS
            __builtin_amdgcn_global_load_async_to_lds_b128(
                (__attribute__((address_space(1))) vsi4*)(void*)gp,
                (__attribute__((address_space(3))) vsi4*)(void*)lp, 0, 0);
#else
            const float4 v = *(const float4*)gp;
            lp[0] = v.x; lp[1] = v.y; lp[2] = v.z; lp[3] = v.w;
#endif
        }
    };

    v8f c0 = {};                                    // tile0: rows wave*32+0..15
    v8f c1 = {};                                    // tile1: rows wave*32+16..31

    issue_chunk(0, xs[0]);
    for (int chunk = 0; chunk < NCHUNK; ++chunk) {
        async_wait0();           // my async writes for buf[chunk&1] landed
        __syncthreads();         // everyone's landed; all done reading buf^1
        if (chunk + 1 < NCHUNK)
            issue_chunk(chunk + 1, xs[(chunk + 1) & 1]);   // overlap with compute

        const float* cur  = xs[chunk & 1];
        const float* a0row = &cur[(wave * 32 + ecol) * LDS_STRIDE];
        const float* a1row = a0row + 16 * LDS_STRIDE;
        const float* brow  = &We[(size_t)ecol * DIM + chunk * BK];
#pragma unroll
        for (int kk = 0; kk < BK; kk += 4) {
            v2f b;  b.x  = brow[kk + khalf];  b.y  = brow[kk + khalf + 1];
            v2f a0; a0.x = a0row[kk + khalf]; a0.y = a0row[kk + khalf + 1];
            v2f a1; a1.x = a1row[kk + khalf]; a1.y = a1row[kk + khalf + 1];
            // one B fragment feeds two independent WMMAs
            c0 = __builtin_amdgcn_wmma_f32_16x16x4_f32(
                     false, a0, false, b, (short)0, c0, false, false);
            c1 = __builtin_amdgcn_wmma_f32_16x16x4_f32(
                     false, a1, false, b, (short)0, c1, false, false);
        }
    }

    // c{t}[r] = score(token = wave*32 + t*16 + r + (lane>=16 ? 8 : 0), expert = ecol)
    const float scale = 0.03125f;                   // 1024^-0.5
    const float bias  = be[ecol];
    float lane_score = 0.0f;                        // sum of w over my tokens
    float lane_aux   = 0.0f;                        // sum of w*log(w+1e-9)

    v8f cc[2] = {c0, c1};
#pragma unroll
    for (int t = 0; t < 2; ++t) {
#pragma unroll
        for (int r = 0; r < 8; ++r) {
            float s = (cc[t][r] + bias) * scale;
            if (ecol >= NEXP) s = -__builtin_inff();    // mask padded experts
            // softmax across the 16-lane expert group
            float m = s;
#pragma unroll
            for (int off = 1; off < 16; off <<= 1)
                m = fmaxf(m, __shfl_xor(m, off, 32));
            const float p = __expf(s - m);              // pads -> exp(-inf)=0
            float den = p;
#pragma unroll
            for (int off = 1; off < 16; off <<= 1)
                den += __shfl_xor(den, off, 32);
            const float w = p / den;
            lane_score += w;
            if (ecol < NEXP) lane_aux += w * __logf(w + 1e-9f);
        }
    }

    // lane L and L^16 share the same expert (different token halves)
    lane_score += __shfl_xor(lane_score, 16, 32);
#pragma unroll
    for (int off = 1; off < 32; off <<= 1)
        lane_aux += __shfl_xor(lane_aux, off, 32);

    if (lane < NEXP_PAD) wpart[wave][lane] = lane_score;
    if (lane == 0)       apart[wave] = lane_aux;
    __syncthreads();

    // deterministic block reduction (wave 0)
    if (tid < NEXP) {
        float a = 0.0f;
#pragma unroll
        for (int w8 = 0; w8 < WAVES_PER_BLK; ++w8) a += wpart[w8][tid];
        block_scores[blockIdx.x * NEXP + tid] = a;
    }
    if (tid == 0) {
        float a = 0.0f;
#pragma unroll
        for (int w8 = 0; w8 < WAVES_PER_BLK; ++w8) a += apart[w8];
        block_aux[blockIdx.x] = a;
    }
}

// ---------------------------------------------------------------------------
// Kernel 3: per-batch score reduction, top-2, mask2d, router loss.
// One block, 64 threads (one per batch).
// ---------------------------------------------------------------------------
__global__ __launch_bounds__(64) void router_finalize(
    const float* __restrict__ block_scores, const float* __restrict__ block_aux,
    float* __restrict__ mask2d, float* __restrict__ out)
{
    __shared__ float sm2d[BATCH * NEXP];
    const int b = threadIdx.x;

    float sc[NEXP];
#pragma unroll
    for (int e = 0; e < NEXP; ++e) {
        float a = 0.0f;
        for (int j = 0; j < BLOCKS_PER_BATCH; ++j)
            a += block_scores[(b * BLOCKS_PER_BATCH + j) * NEXP + e];
        sc[e] = a * (1.0f / (float)SEQ);            // mean over S
    }
    // top-2, first index wins ties (matches lax.top_k)
    int i0 = 0;
#pragma unroll
    for (int e = 1; e < NEXP; ++e) if (sc[e] > sc[i0]) i0 = e;
    int i1 = (i0 == 0) ? 1 : 0;
#pragma unroll
    for (int e = 0; e < NEXP; ++e)
        if (e != i0 && sc[e] > sc[i1]) i1 = e;

#pragma unroll
    for (int e = 0; e < NEXP; ++e) {
        const float m = (e == i0 || e == i1) ? 1.0f : 0.0f;
        mask2d[b * NEXP + e] = m;
        sm2d[b * NEXP + e]   = m;
    }
    out[MASK_ELEMS + b * TOPK + 0] = (float)i0;
    out[MASK_ELEMS + b * TOPK + 1] = (float)i1;
    __syncthreads();

    if (b == 0) {
        // kl = sum_e ideal*(log ideal - log usage) / E ; ideal = 1/8
        float kl = 0.0f;
#pragma unroll
        for (int e = 0; e < NEXP; ++e) {
            float cnt = 0.0f;
            for (int bb = 0; bb < BATCH; ++bb) cnt += sm2d[bb * NEXP + e];
            const float usage = cnt * (1.0f / (float)BATCH);
            kl += 0.125f * (__logf(0.125f) - __logf(usage));
        }
        kl *= (1.0f / (float)NEXP);

        float aux = 0.0f;
        for (int j = 0; j < NBLOCKS_MAIN; ++j) aux += block_aux[j];
        aux *= (1.0f / (float)MASK_ELEMS);          // mean over B*S*E

        out[MASK_ELEMS + BATCH * TOPK] = 1e-3f * kl + 1e-3f * aux;
    }
}

// ---------------------------------------------------------------------------
// Kernel 4: broadcast mask2d [B,E] -> mask [B,S,E]
// ---------------------------------------------------------------------------
__global__ __launch_bounds__(256) void router_write_mask(
    const float* __restrict__ mask2d, float* __restrict__ out)
{
    const int i = blockIdx.x * 256 + threadIdx.x;
    if (i < MASK_ELEMS) {
        const int b = i >> 13;                      // / (S*E) = 8192
        const int e = i & (NEXP - 1);
        out[i] = mask2d[b * NEXP + e];
    }
}

// ---------------------------------------------------------------------------
extern "C" void kernel_launch(void* const* d_in, const int* in_sizes, int n_in,
                              void* d_out, int out_size, void* d_ws, size_t ws_size,
                              hipStream_t stream) {
    const float* x       = (const float*)d_in[0];   // [B,S,D]
    const float* Wq      = (const float*)d_in[1];   // [D,D]
    const float* bq      = (const float*)d_in[2];   // [D]
    const float* key_emb = (const float*)d_in[3];   // [E,D]
    float* out = (float*)d_out;

    float* ws = (float*)d_ws;
    float* We           = ws;                        // 16*1024 = 16384
    float* be           = ws + 16384;                // 16
    float* block_scores = ws + 16400;                // 256*8 = 2048
    float* block_aux    = ws + 18448;                // 256
    float* mask2d       = ws + 18704;                // 512

    router_prep<<<DIM / 256, 256, 0, stream>>>(Wq, bq, key_emb, We, be);
    router_main<<<NBLOCKS_MAIN, THREADS, 0, stream>>>(x, We, be, block_scores, block_aux);
    router_finalize<<<1, BATCH, 0, stream>>>(block_scores, block_aux, mask2d, out);
    router_write_mask<<<(MASK_ELEMS + 255) / 256, 256, 0, stream>>>(mask2d, out);
}